// HypAgg_31413390803679
// MI455X (gfx1250) — compile-verified
//
#include <hip/hip_runtime.h>
#include <math.h>

// ---- problem constants (from reference) ----
#define DFEAT 64
#define MIN_NORM 1e-15f
#define BALL_EPS 4e-3f
#define SQRT_C 1.0f   // c = 1.0

typedef float v2f __attribute__((ext_vector_type(2)));
typedef float v8f __attribute__((ext_vector_type(8)));

// Wave-level squared-row-norms for 16 rows x 64 cols via V_WMMA_F32_16X16X4_F32.
// A-fragment of X(16x4) and B-fragment of X^T(4x16) are register-identical
// (lane%16 = M/N index, {VGPR, lane>=16} = K index), so wmma(a,a,acc) = X X^T.
// Accumulating 16 chunks of K=4 covers K=64; diag(acc) = squared norms.
// Returns this lane's norm^2 for row m = lane%16, and fills a[16] fragments.
__device__ inline float gram_norm2(const float* __restrict__ rowptr,
                                   int lane, int h, v2f a[16]) {
  v8f acc = {};
#pragma unroll
  for (int kc = 0; kc < 16; ++kc) {
    // lane m (=lane%16), half h: holds X[m][4*kc + 2*h .. +1]
    a[kc] = *(const v2f*)(rowptr + kc * 4 + h * 2);
  }
#pragma unroll
  for (int kc = 0; kc < 16; ++kc) {
    acc = __builtin_amdgcn_wmma_f32_16x16x4_f32(
        /*neg_a=*/false, a[kc], /*neg_b=*/false, a[kc],
        /*c_mod=*/(short)0, acc, /*reuse_a=*/false, /*reuse_b=*/false);
  }
  // C/D layout: VGPR v, lanes 0-15 -> (M=v, N=lane); lanes 16-31 -> (M=v+8, N=lane-16).
  // diag(m) lives at VGPR m&7, lane (m<8 ? m : m+16).
  int m = lane & 15;
  int srcLane = (m < 8) ? m : (m + 16);
  float n2 = 0.0f;
#pragma unroll
  for (int v = 0; v < 8; ++v) {
    float t = __shfl(acc[v], srcLane, 32);
    if ((m & 7) == v) n2 = t;
  }
  return n2;
}

// ---------------- kernel 1: x_t = logmap0(x) ----------------
__global__ __launch_bounds__(256) void k_logmap0(const float* __restrict__ x,
                                                 float* __restrict__ xt, int n) {
  int lane = threadIdx.x & 31;
  int wave = blockIdx.x * 8 + (threadIdx.x >> 5);
  int base = wave * 16;
  if (base >= n) return;                    // whole-wave uniform exit (EXEC stays full)
  int m = lane & 15;
  int h = lane >> 4;
  int row = base + m;
  int rowc = row < n ? row : (n - 1);       // clamp loads for ragged tail

  v2f a[16];
  float n2 = gram_norm2(x + (size_t)rowc * DFEAT, lane, h, a);

  float norm = fmaxf(sqrtf(fmaxf(n2, 0.0f)), MIN_NORM);
  float t = fminf(SQRT_C * norm, 1.0f - 1e-7f);
  float art = 0.5f * (log1pf(t) - log1pf(-t));       // artanh
  float scale = art / (SQRT_C * norm);

  if (row < n) {
    float* o = xt + (size_t)row * DFEAT;
#pragma unroll
    for (int kc = 0; kc < 16; ++kc) {
      v2f w; w.x = a[kc].x * scale; w.y = a[kc].y * scale;
      *(v2f*)(o + kc * 4 + h * 2) = w;
    }
  }
}

// ---------------- kernel 2: zero the accumulator ----------------
__global__ __launch_bounds__(256) void k_zero4(float4* __restrict__ p, int n4) {
  int i = blockIdx.x * 256 + threadIdx.x;
  if (i < n4) p[i] = make_float4(0.f, 0.f, 0.f, 0.f);
}

// ---------------- kernel 3: COO scatter  sup[r] += val * x_t[c] ----------------
// One wave per edge: lanes 0..31 each handle a float2 of the 64-wide row.
// Edge metadata loads are wave-uniform (scalarized); gathers/atomics hit L2
// (16 MB working set << 192 MB L2).
__global__ __launch_bounds__(256) void k_scatter(const float* __restrict__ xt,
                                                 const float* __restrict__ ev,
                                                 const int* __restrict__ rows,
                                                 const int* __restrict__ cols,
                                                 float* __restrict__ sup, int ne) {
  int gid = blockIdx.x * 256 + threadIdx.x;
  int e = gid >> 5;
  if (e >= ne) return;
  int lane = gid & 31;
  int r = rows[e];
  int c = cols[e];
  float v = ev[e];
  v2f g = *(const v2f*)(xt + (size_t)c * DFEAT + lane * 2);
  float* dst = sup + (size_t)r * DFEAT + lane * 2;
  atomicAdd(dst, g.x * v);
  atomicAdd(dst + 1, g.y * v);
}

// ---------------- kernel 4: out = proj(expmap0(support)) (in place) ----------------
__global__ __launch_bounds__(256) void k_expproj(float* __restrict__ u, int n) {
  int lane = threadIdx.x & 31;
  int wave = blockIdx.x * 8 + (threadIdx.x >> 5);
  int base = wave * 16;
  if (base >= n) return;
  int m = lane & 15;
  int h = lane >> 4;
  int row = base + m;
  int rowc = row < n ? row : (n - 1);

  v2f a[16];
  float n2 = gram_norm2(u + (size_t)rowc * DFEAT, lane, h, a);

  float un = fmaxf(sqrtf(fmaxf(n2, 0.0f)), MIN_NORM);
  float th = tanhf(SQRT_C * un);
  float factor = th / (SQRT_C * un);             // expmap0 scale
  float ynorm = fmaxf(th / SQRT_C, MIN_NORM);    // |expmap0(u)| analytically
  float maxnorm = (1.0f - BALL_EPS) / SQRT_C;
  float s = factor * fminf(1.0f, maxnorm / ynorm);

  if (row < n) {
    float* o = u + (size_t)row * DFEAT;
#pragma unroll
    for (int kc = 0; kc < 16; ++kc) {
      v2f w; w.x = a[kc].x * s; w.y = a[kc].y * s;
      *(v2f*)(o + kc * 4 + h * 2) = w;
    }
  }
}

extern "C" void kernel_launch(void* const* d_in, const int* in_sizes, int n_in,
                              void* d_out, int out_size, void* d_ws, size_t ws_size,
                              hipStream_t stream) {
  const float* x    = (const float*)d_in[0];   // [N, 64]
  const float* ev   = (const float*)d_in[1];   // [E]
  const int*   rows = (const int*)d_in[2];     // [E]
  const int*   cols = (const int*)d_in[3];     // [E]
  float* out = (float*)d_out;                  // [N, 64], doubles as support accumulator
  float* xt  = (float*)d_ws;                   // [N, 64] tangent-space scratch

  int n  = in_sizes[0] / DFEAT;
  int ne = in_sizes[1];

  int waves   = (n + 15) / 16;
  int rblocks = (waves + 7) / 8;              // 8 waves (256 threads) per block

  // 1) tangent map
  k_logmap0<<<rblocks, 256, 0, stream>>>(x, xt, n);

  // 2) zero accumulator (harness poisons d_out)
  int n4 = (n * DFEAT) / 4;
  k_zero4<<<(n4 + 255) / 256, 256, 0, stream>>>((float4*)out, n4);

  // 3) SpMM scatter: wave per edge
  int sthreads = ne * 32;
  k_scatter<<<(sthreads + 255) / 256, 256, 0, stream>>>(xt, ev, rows, cols, out, ne);

  // 4) expmap0 + proj, in place on d_out
  k_expproj<<<rblocks, 256, 0, stream>>>(out, n);
}